// CustomOp_28346784153783
// MI455X (gfx1250) — compile-verified
//
#include <hip/hip_runtime.h>
#include <stdint.h>

#define DCOLS 64
#define TILE_ROWS 16
#define THREADS 64
#define CHUNKS ((TILE_ROWS * DCOLS * 4) / 16)   // 256 x 16B chunks per tile
#define CHUNKS_PER_THREAD (CHUNKS / THREADS)    // 4 async b128 ops / thread / tile

typedef int v4i __attribute__((ext_vector_type(4)));
typedef __attribute__((address_space(1))) v4i gv4i;   // global
typedef __attribute__((address_space(3))) v4i lv4i;   // LDS

#ifdef __has_builtin
#if __has_builtin(__builtin_amdgcn_global_load_async_to_lds_b128)
#define HAVE_ASYNC_BUILTIN 1
#endif
#if __has_builtin(__builtin_amdgcn_s_wait_asynccnt)
#define HAVE_WAIT_BUILTIN 1
#endif
#endif

// Streamed-once data: TH=NT (cpol=1) so the 1 GiB of x doesn't evict the
// L2-resident index array (32 MB, hit ~44x by the binary searches).
__device__ __forceinline__ void async_copy_b128_nt(const void* gsrc, void* ldst) {
#ifdef HAVE_ASYNC_BUILTIN
    __builtin_amdgcn_global_load_async_to_lds_b128(
        (gv4i*)gsrc, (lv4i*)ldst, /*offset=*/0, /*cpol=*/1);
#else
    unsigned loff = (unsigned)(unsigned long long)
        (__attribute__((address_space(3))) char*)ldst;
    unsigned long long ga = (unsigned long long)gsrc;
    asm volatile("global_load_async_to_lds_b128 %0, %1, off th:TH_LOAD_NT"
                 :: "v"(loff), "v"(ga) : "memory");
#endif
}

__device__ __forceinline__ void wait_async_le4() {
#ifdef HAVE_WAIT_BUILTIN
    __builtin_amdgcn_s_wait_asynccnt(4);
#else
    asm volatile("s_wait_asynccnt 0x4" ::: "memory");
#endif
}

__device__ __forceinline__ void wait_async_0() {
#ifdef HAVE_WAIT_BUILTIN
    __builtin_amdgcn_s_wait_asynccnt(0);
#else
    asm volatile("s_wait_asynccnt 0x0" ::: "memory");
#endif
}

__device__ __forceinline__ int lower_bound_i64(const long long* __restrict__ idx,
                                               int n, long long key) {
    int l = 0, r = n;
    while (l < r) {
        int m = (l + r) >> 1;
        if (idx[m] < key) l = m + 1; else r = m;
    }
    return l;
}

// One block per segment. Sorted index => contiguous row range per segment.
__global__ __launch_bounds__(THREADS) void segmax_kernel(
    const float* __restrict__ x, const long long* __restrict__ idx,
    float* __restrict__ out, int n) {
    __shared__ float buf[2][TILE_ROWS * DCOLS];   // 2 x 4KB double buffer
    __shared__ int sh_lo, sh_hi;

    const int s = blockIdx.x;
    const int t = threadIdx.x;

    // Run the two lower-bound searches concurrently, one per wave, to halve
    // the dependent scalar-load latency chain.
    if (t == 0)  sh_lo = lower_bound_i64(idx, n, (long long)s);
    if (t == 32) sh_hi = lower_bound_i64(idx, n, (long long)s + 1);
    __syncthreads();

    const int lo = sh_lo, hi = sh_hi;
    const int nrows = hi - lo;
    const int full = nrows >> 4;   // full 16-row tiles

    float m = -__builtin_inff();

    // Prime the pipeline: async-stage tile 0 into buf[0]
    if (full > 0) {
        const char* gbase = (const char*)(x + (size_t)lo * DCOLS);
        char* lbase = (char*)&buf[0][0];
#pragma unroll
        for (int c = 0; c < CHUNKS_PER_THREAD; ++c) {
            const int chunk = t + c * THREADS;
            async_copy_b128_nt(gbase + (size_t)chunk * 16, lbase + chunk * 16);
        }
    }

    for (int k = 0; k < full; ++k) {
        if (k + 1 < full) {
            // Issue next tile, then wait leaving the 4 just-issued ops
            // outstanding: in-order async completion => tile k is resident.
            const char* gbase =
                (const char*)(x + (size_t)(lo + (k + 1) * TILE_ROWS) * DCOLS);
            char* lbase = (char*)&buf[(k + 1) & 1][0];
#pragma unroll
            for (int c = 0; c < CHUNKS_PER_THREAD; ++c) {
                const int chunk = t + c * THREADS;
                async_copy_b128_nt(gbase + (size_t)chunk * 16, lbase + chunk * 16);
            }
            wait_async_le4();
        } else {
            wait_async_0();
        }
        __syncthreads();   // both waves' async writes now visible

        const float* bb = &buf[k & 1][0];
#pragma unroll
        for (int r = 0; r < TILE_ROWS; ++r)
            m = fmaxf(m, bb[r * DCOLS + t]);   // 64 banks, stride 64 -> conflict-free

        __syncthreads();   // done reading before this buffer is overwritten
    }

    // Tail rows (< 16): plain coalesced global loads, also non-temporal
    for (int r = lo + full * TILE_ROWS; r < hi; ++r)
        m = fmaxf(m, __builtin_nontemporal_load(&x[(size_t)r * DCOLS + t]));

    // torch_scatter semantics: empty segment -> 0, else true max (can be negative)
    out[(size_t)s * DCOLS + t] = (nrows > 0) ? m : 0.0f;
}

extern "C" void kernel_launch(void* const* d_in, const int* in_sizes, int n_in,
                              void* d_out, int out_size, void* d_ws, size_t ws_size,
                              hipStream_t stream) {
    const float* x = (const float*)d_in[0];
    const long long* idx = (const long long*)d_in[1];
    float* out = (float*)d_out;
    const int n = in_sizes[1];                 // 4194304 rows
    const int num_segments = out_size / DCOLS; // 65536

    segmax_kernel<<<num_segments, THREADS, 0, stream>>>(x, idx, out, n);
}